// HGINLayer_21912923144305
// MI455X (gfx1250) — compile-verified
//
#include <hip/hip_runtime.h>
#include <hip/hip_bf16.h>

typedef float v2f __attribute__((ext_vector_type(2)));
typedef float v8f __attribute__((ext_vector_type(8)));

// ---------------------------------------------------------------- zero
__global__ void k_zero_f32(float* __restrict__ p, long n) {
    long i = (long)blockIdx.x * blockDim.x + threadIdx.x;
    long stride = (long)gridDim.x * blockDim.x;
    for (; i < n; i += stride) p[i] = 0.0f;
}

// ---------------------------------------------------------------- h = (1+eps)*x_t
__global__ void k_init_h(const float* __restrict__ x, const float* __restrict__ eps,
                         int eps_idx, float* __restrict__ h, long n) {
    float s = 1.0f + eps[eps_idx];
    long i = (long)blockIdx.x * blockDim.x + threadIdx.x;
    long stride = (long)gridDim.x * blockDim.x;
    for (; i < n; i += stride) h[i] = s * x[i];
}

// ---------------------------------------------------------------- scatter-add along edges
__global__ void k_scatter_add(const float* __restrict__ xs, const int* __restrict__ src,
                              const int* __restrict__ dst, float* __restrict__ h, int n_edges) {
    long total = (long)n_edges * 128;
    long i = (long)blockIdx.x * blockDim.x + threadIdx.x;
    long stride = (long)gridDim.x * blockDim.x;
    for (; i < total; i += stride) {
        int e = (int)(i >> 7);
        int d = (int)(i & 127);
        atomicAdd(&h[(long)dst[e] * 128 + d], xs[(long)src[e] * 128 + d]);
    }
}

// ---------------------------------------------------------------- Y = A(16m x 128) @ W(128x128) + bias  via V_WMMA_F32_16X16X4_F32
__global__ void __launch_bounds__(256)
k_gemm_bias_wmma(const float* __restrict__ A, const float* __restrict__ W,
                 const float* __restrict__ bias, float* __restrict__ Y, int n_tiles) {
    __shared__ float lds[16 * 128];
    const int tid   = threadIdx.x;
    const int wave  = tid >> 5;          // 0..7 -> 16-column strip
    const int lane  = tid & 31;
    const int m     = lane & 15;         // A row / output column-in-strip
    const int khalf = (lane >> 4) * 2;   // 0 for lanes 0-15, 2 for lanes 16-31
    const int n     = wave * 16 + m;     // global output column

    // Preload B fragments for all 32 K-steps (K=4 each): lane holds W[kb][n], W[kb+1][n]
    v2f bfrag[32];
#pragma unroll
    for (int k = 0; k < 32; ++k) {
        int kb = k * 4 + khalf;
        bfrag[k].x = W[kb * 128 + n];
        bfrag[k].y = W[(kb + 1) * 128 + n];
    }
    const float bval = bias[n];

    for (int tile = blockIdx.x; tile < n_tiles; tile += gridDim.x) {
        long rowbase = (long)tile * 16;
        __syncthreads();
#pragma unroll
        for (int i = 0; i < 8; ++i) {           // 16x128 A tile, coalesced
            int idx = tid + i * 256;
            lds[idx] = A[rowbase * 128 + idx];
        }
        __syncthreads();

        v8f c = {};
#pragma unroll
        for (int k = 0; k < 32; ++k) {
            v2f a = *(const v2f*)&lds[m * 128 + k * 4 + khalf];
            c = __builtin_amdgcn_wmma_f32_16x16x4_f32(
                    /*neg_a=*/false, a, /*neg_b=*/false, bfrag[k],
                    /*c_mod=*/(short)0, c, /*reuse_a=*/false, /*reuse_b=*/false);
        }

        const int rowoff = (lane < 16) ? 0 : 8;  // C VGPR r: M=r (low lanes) / r+8 (high lanes)
#pragma unroll
        for (int r = 0; r < 8; ++r) {
            Y[(rowbase + rowoff + r) * 128 + n] = c[r] + bval;
        }
    }
}

// ---------------------------------------------------------------- per-column sum / sum-of-squares
__global__ void k_col_stats(const float* __restrict__ Y, int n_rows, float* __restrict__ sums) {
    int c = threadIdx.x;                  // blockDim.x == 128
    float s = 0.0f, ss = 0.0f;
    for (int r = blockIdx.x; r < n_rows; r += gridDim.x) {
        float v = Y[(long)r * 128 + c];
        s += v;
        ss += v * v;
    }
    atomicAdd(&sums[c], s);
    atomicAdd(&sums[128 + c], ss);
}

// ---------------------------------------------------------------- Z = relu(bn(Y))
__global__ void k_bn_relu(const float* __restrict__ Y, const float* __restrict__ sums,
                          const float* __restrict__ g, const float* __restrict__ be,
                          float* __restrict__ Z, long n_elems, float inv_n) {
    long i = (long)blockIdx.x * blockDim.x + threadIdx.x;
    long stride = (long)gridDim.x * blockDim.x;
    for (; i < n_elems; i += stride) {
        int c = (int)(i & 127);
        float mean = sums[c] * inv_n;
        float var  = fmaf(-mean, mean, sums[128 + c] * inv_n);   // biased var
        float z = g[c] * (Y[i] - mean) * rsqrtf(var + 1e-5f) + be[c];
        Z[i] = fmaxf(z, 0.0f);
    }
}

// ---------------------------------------------------------------- out += relu(bn(Y))
__global__ void k_bn_relu_acc(const float* __restrict__ Y, const float* __restrict__ sums,
                              const float* __restrict__ g, const float* __restrict__ be,
                              float* __restrict__ out, long n_elems, float inv_n) {
    long i = (long)blockIdx.x * blockDim.x + threadIdx.x;
    long stride = (long)gridDim.x * blockDim.x;
    for (; i < n_elems; i += stride) {
        int c = (int)(i & 127);
        float mean = sums[c] * inv_n;
        float var  = fmaf(-mean, mean, sums[128 + c] * inv_n);
        float z = g[c] * (Y[i] - mean) * rsqrtf(var + 1e-5f) + be[c];
        out[i] += fmaxf(z, 0.0f);
    }
}

// ================================================================ host orchestration
static inline int imin(int a, int b) { return a < b ? a : b; }

extern "C" void kernel_launch(void* const* d_in, const int* in_sizes, int n_in,
                              void* d_out, int out_size, void* d_ws, size_t ws_size,
                              hipStream_t stream) {
    (void)in_sizes; (void)n_in; (void)out_size; (void)ws_size;

    const float* x[3] = {(const float*)d_in[0],   // operation (100000 x 128)
                         (const float*)d_in[1],   // machine   (20000 x 128)
                         (const float*)d_in[2]};  // job       (20000 x 128)
    const long N[3] = {100000, 20000, 20000};
    const long out_off[3] = {0, 100000L * 128, 120000L * 128};

    const float* params = (const float*)d_in[12];
    const float* eps    = (const float*)d_in[13];
    float* out = (float*)d_out;

    // workspace layout
    char* ws = (char*)d_ws;
    float* bufA = (float*)ws;                        // 100000*128 fp32 = 51.2 MB
    float* bufB = (float*)(ws + 51200000L);          // 51.2 MB
    float* sums = (float*)(ws + 102400000L);         // 256 fp32

    // pytree-flatten order: outer keys sorted (job, machine, operation);
    // inner keys sorted (W1, W2, b1, b2, be1, be2, g1, g2); block = 2*16384 + 6*128
    const long BLK = 33536;
    const int src_block[3] = {2, 1, 0};   // source type op/ma/jo -> sorted position

    const int n_edges = 200000;

    // zero outputs once (harness poisons d_out)
    k_zero_f32<<<4096, 256, 0, stream>>>(out, 140000L * 128);

    for (int i = 0; i < 9; ++i) {
        const int s = i / 3;              // source type: 0=op 1=ma 2=jo
        const int t = i % 3;              // target type
        const long nt = N[t];
        const long ne = nt * 128;
        const float inv_n = 1.0f / (float)nt;
        const int n_tiles = (int)(nt / 16);          // 100000,20000 both %16==0
        const int gemm_grid = imin(n_tiles, 2048);

        const float* P   = params + src_block[s] * BLK;
        const float* W1  = P;
        const float* W2  = P + 16384;
        const float* b1  = P + 32768;
        const float* b2  = P + 32896;
        const float* be1 = P + 33024;
        const float* be2 = P + 33152;
        const float* g1  = P + 33280;
        const float* g2  = P + 33408;

        const int* ei = (const int*)d_in[3 + i];     // [2, 200000] row-major

        // h = (1+eps_i) * x_t  (self term), then scatter-add gathered source features
        k_init_h<<<4096, 256, 0, stream>>>(x[t], eps, i, bufA, ne);
        k_scatter_add<<<8192, 256, 0, stream>>>(x[s], ei, ei + n_edges, bufA, n_edges);

        // layer 1: y1 = h @ W1 + b1 ; bn + relu -> bufA
        k_gemm_bias_wmma<<<gemm_grid, 256, 0, stream>>>(bufA, W1, b1, bufB, n_tiles);
        k_zero_f32<<<1, 256, 0, stream>>>(sums, 256);
        k_col_stats<<<512, 128, 0, stream>>>(bufB, (int)nt, sums);
        k_bn_relu<<<2048, 256, 0, stream>>>(bufB, sums, g1, be1, bufA, ne, inv_n);

        // layer 2: y2 = z1 @ W2 + b2 ; bn + relu accumulated into output segment
        k_gemm_bias_wmma<<<gemm_grid, 256, 0, stream>>>(bufA, W2, b2, bufB, n_tiles);
        k_zero_f32<<<1, 256, 0, stream>>>(sums, 256);
        k_col_stats<<<512, 128, 0, stream>>>(bufB, (int)nt, sums);
        k_bn_relu_acc<<<2048, 256, 0, stream>>>(bufB, sums, g2, be2, out + out_off[t], ne, inv_n);
    }
}